// Normalized_resampleGraph_expand_37709812859475
// MI455X (gfx1250) — compile-verified
//
#include <hip/hip_runtime.h>
#include <stdint.h>

#define BATCH 4
#define NV    10000
#define NF    20000
#define CUT   8
#define NS    80000
#define CH    128

typedef __attribute__((ext_vector_type(2))) float v2f;
typedef __attribute__((ext_vector_type(8))) float v8f;

// Exact spelling from the compiler diagnostic: int4 via __vector_size__.
typedef int v4i_t __attribute__((__vector_size__(4 * sizeof(int))));
typedef __attribute__((address_space(1))) v4i_t gv4i;  // global int4
typedef __attribute__((address_space(3))) v4i_t lv4i;  // LDS int4

// Async global -> LDS copy of 16 bytes per lane (CDNA5 ASYNCcnt path).
__device__ __forceinline__ void async_copy_b128(const void* gsrc, void* ldst) {
#if __has_builtin(__builtin_amdgcn_global_load_async_to_lds_b128)
  __builtin_amdgcn_global_load_async_to_lds_b128((gv4i*)gsrc, (lv4i*)ldst, 0, 0);
#else
  unsigned la = (unsigned)(uintptr_t)(lv4i*)ldst;
  unsigned long long ga = (unsigned long long)(uintptr_t)gsrc;
  asm volatile("global_load_async_to_lds_b128 %0, %1, off" :: "v"(la), "v"(ga) : "memory");
#endif
}

__device__ __forceinline__ void wait_async0() {
#if __has_builtin(__builtin_amdgcn_s_wait_asynccnt)
  __builtin_amdgcn_s_wait_asynccnt(0);
#else
  asm volatile("s_wait_asynccnt 0x0" ::: "memory");
#endif
}

// One block = one (b, v) pair; wave k (0..7) produces out[b, v, k, 0:128].
__global__ __launch_bounds__(256) void resample_graph_expand_kernel(
    const float* __restrict__ xf,   // (4, 10000, 128)
    const float* __restrict__ Bw,   // (4, 80000, 3)
    const int*   __restrict__ xg,   // (10000, 8)
    const int*   __restrict__ Fc,   // (20000, 3)
    const int*   __restrict__ Iv,   // (80000,)
    float*       __restrict__ out)  // (4, 10000, 8, 128)
{
  __shared__ float smem[8 * 3 * CH];  // 12 KB: 3 gathered rows per wave

  const int bv   = blockIdx.x;            // 0 .. BATCH*NV-1
  const int b    = bv / NV;
  const int v    = bv - b * NV;
  const int lane = threadIdx.x & 31;
  const int k    = __builtin_amdgcn_readfirstlane((int)(threadIdx.x >> 5));

  // Wave-uniform index chain (scalar loads).
  const int g  = xg[v * CUT + k];
  const int f  = Iv[g];
  const int i0 = Fc[f * 3 + 0];
  const int i1 = Fc[f * 3 + 1];
  const int i2 = Fc[f * 3 + 2];
  const long bw = ((long)b * NS + g) * 3;
  const float w0 = Bw[bw + 0];
  const float w1 = Bw[bw + 1];
  const float w2 = Bw[bw + 2];

  const long fb = (long)b * NV;
  const float* r0 = xf + (fb + i0) * CH;
  const float* r1 = xf + (fb + i1) * CH;
  const float* r2 = xf + (fb + i2) * CH;

  // Stage 3x512B gathered feature rows into this wave's LDS slab (async).
  float* lbase = &smem[k * (3 * CH)];
  async_copy_b128(r0 + lane * 4, lbase + 0 * CH + lane * 4);
  async_copy_b128(r1 + lane * 4, lbase + 1 * CH + lane * 4);
  async_copy_b128(r2 + lane * 4, lbase + 2 * CH + lane * 4);
  wait_async0();  // per-wave counter; no block barrier needed

  const bool lo = lane < 16;
  const int  n  = lane & 15;

  // A (16x4 f32): VGPR0 = K0 | K2 halves, VGPR1 = K1 | K3(pad=0).
  // All 16 M-rows get the same weights -> all D rows equal the answer.
  v2f A;
  A.x = lo ? w0 : w2;
  A.y = lo ? w1 : 0.0f;

  float res[8];
#pragma unroll
  for (int i = 0; i < 8; ++i) {
    // B-matrix (4x16 f32) for channel slice [16i, 16i+16): mirror of A halves.
    v2f Bm;
    Bm.x = lbase[(lo ? 0 : 2) * CH + i * 16 + n];
    float t = lbase[1 * CH + i * 16 + n];
    Bm.y = lo ? t : 0.0f;  // keep K=3 row finite (0*NaN would poison WMMA)

    v8f c = {};
    v8f d = __builtin_amdgcn_wmma_f32_16x16x4_f32(
        /*neg_a=*/false, A, /*neg_b=*/false, Bm,
        /*c_mod=*/(short)0, c, /*reuse_a=*/false, /*reuse_b=*/false);
    res[i] = d[0];  // VGPR0: lanes 0-15 = (M=0, N=n), lanes 16-31 = (M=8, N=n)
  }

  // out[b,v,k,c] = weighted_sum - self[c]; pair slices -> 128B contiguous stores.
  const float* self = xf + (fb + v) * CH;
  float* orow = out + ((long)bv * CUT + k) * CH;
#pragma unroll
  for (int p = 0; p < 4; ++p) {
    const int c = 32 * p + lane;             // lo: slice 2p, hi: slice 2p+1
    const float val = lo ? res[2 * p] : res[2 * p + 1];
    orow[c] = val - self[c];
  }
}

extern "C" void kernel_launch(void* const* d_in, const int* in_sizes, int n_in,
                              void* d_out, int out_size, void* d_ws, size_t ws_size,
                              hipStream_t stream) {
  const float* xf = (const float*)d_in[0];  // x_features
  const float* Bw = (const float*)d_in[1];  // B
  const int*   xg = (const int*)d_in[2];    // x_graph
  const int*   Fc = (const int*)d_in[3];    // F
  const int*   Iv = (const int*)d_in[4];    // I
  float* out = (float*)d_out;

  dim3 grid(BATCH * NV);
  dim3 block(256);
  hipLaunchKernelGGL(resample_graph_expand_kernel, grid, block, 0, stream,
                     xf, Bw, xg, Fc, Iv, out);
}